// GenAttentionAggregation_72086731096452
// MI455X (gfx1250) — compile-verified
//
#include <hip/hip_runtime.h>
#include <math.h>

#define N_NODES 131072
#define DIM     512
#define SEGS    4096
#define EPS_F   1e-16f

typedef __attribute__((ext_vector_type(2))) float v2f;
typedef __attribute__((ext_vector_type(8))) float v8f;

// float atomic max via integer-ordering trick (works for any sign, init = -inf)
__device__ __forceinline__ void atomicMaxFloat(float* addr, float value) {
  if (value >= 0.0f)
    atomicMax((int*)addr, __float_as_int(value));
  else
    atomicMin((unsigned int*)addr, __float_as_uint(value));
}

__global__ void init_kernel(float* __restrict__ Y, float* __restrict__ seg_max,
                            float* __restrict__ denom, float* __restrict__ counts) {
  int i = blockIdx.x * blockDim.x + threadIdx.x;
  if (i < SEGS * DIM) Y[i] = 0.0f;
  if (i < SEGS) {
    seg_max[i] = -__builtin_inff();
    denom[i]   = 0.0f;
    counts[i]  = 0.0f;
  }
}

// one wave32 per row: scores[i] = dot(attention_x[i], W_score) + b_score
// also: segment max (atomic) and segment counts (atomic)
__global__ void scores_kernel(const float* __restrict__ ax, const float* __restrict__ Wscore,
                              const float* __restrict__ bscore, const int* __restrict__ index,
                              float* __restrict__ scores, float* __restrict__ seg_max,
                              float* __restrict__ counts) {
  int row  = (blockIdx.x * blockDim.x + threadIdx.x) >> 5;
  int lane = threadIdx.x & 31;
  if (row >= N_NODES) return;
  const float4* a4 = (const float4*)(ax + (size_t)row * DIM);
  const float4* w4 = (const float4*)Wscore;
  float acc = 0.0f;
#pragma unroll
  for (int j = 0; j < (DIM / 4) / 32; ++j) {
    float4 a = a4[lane + 32 * j];
    float4 w = w4[lane + 32 * j];
    acc += a.x * w.x + a.y * w.y + a.z * w.z + a.w * w.w;
  }
#pragma unroll
  for (int off = 16; off > 0; off >>= 1) acc += __shfl_xor(acc, off, 32);
  if (lane == 0) {
    float s = acc + bscore[0];
    scores[row] = s;
    int seg = index[row];
    atomicMaxFloat(&seg_max[seg], s);
    atomicAdd(&counts[seg], 1.0f);
  }
}

// w = exp(score - segmax); denom += w; Y[seg,:] += w * x[row,:]
// one 128-thread block per row; each thread covers 4 dims (float4 load + 4 f32 atomics into L2)
__global__ void accumulate_kernel(const float* __restrict__ x, const int* __restrict__ index,
                                  const float* __restrict__ scores, const float* __restrict__ seg_max,
                                  float* __restrict__ denom, float* __restrict__ Y) {
  int row = blockIdx.x;
  int t   = threadIdx.x;  // 0..127
  int seg = index[row];
  float w = expf(scores[row] - seg_max[seg]);
  if (t == 0) atomicAdd(&denom[seg], w);
  float4 xv = *(const float4*)(x + (size_t)row * DIM + t * 4);
  float* yrow = Y + (size_t)seg * DIM + t * 4;
  atomicAdd(yrow + 0, w * xv.x);
  atomicAdd(yrow + 1, w * xv.y);
  atomicAdd(yrow + 2, w * xv.z);
  atomicAdd(yrow + 3, w * xv.w);
}

// fp32 WMMA GEMM: [SEGS, DIM] x [DIM, DIM], one wave per 16x16 tile, K steps of 4.
// Fused epilogue: out = (inv*acc + denom*inv*b_emb) * (counts*W_size + b_size), inv = 1/(denom+EPS)
__global__ void wmma_gemm_epilogue(const float* __restrict__ Y, const float* __restrict__ W,
                                   const float* __restrict__ b_emb, const float* __restrict__ Wsz,
                                   const float* __restrict__ bsz,
                                   const float* __restrict__ denom, const float* __restrict__ counts,
                                   float* __restrict__ out) {
  const int NT = DIM / 16;                          // 32 tiles along n
  int tile = blockIdx.x * 4 + (threadIdx.x >> 5);   // 4 waves per block
  int lane = threadIdx.x & 31;
  int half = lane >> 4;                             // 0: K{0,1}, 1: K{2,3}
  int l    = lane & 15;
  int tm = tile / NT;
  int tn = tile % NT;

  // A: 16x4 fp32 tile of Y. Lane l<16 holds row l, K+{0,1}; lane>=16 holds row l, K+{2,3}.
  const float* Arow = Y + (size_t)(tm * 16 + l) * DIM + half * 2;
  // B: 4x16 fp32 tile of W (row-major [K, N]). Lane l<16: rows K+{0,1}, col n0+l; lane>=16: rows K+{2,3}.
  const float* Bcol = W + (size_t)(half * 2) * DIM + tn * 16 + l;

  v8f c = {};
#pragma unroll 4
  for (int k = 0; k < DIM; k += 4) {
    v2f a = *(const v2f*)(Arow + k);                // Y[m, k+half*2 .. +1]
    v2f b;
    b.x = Bcol[(size_t)k * DIM];                    // W[k+half*2    , n]
    b.y = Bcol[(size_t)(k + 1) * DIM];              // W[k+half*2 + 1, n]
    c = __builtin_amdgcn_wmma_f32_16x16x4_f32(false, a, false, b, (short)0, c, false, false);
  }

  // C/D layout: lane gives N = l (+0/+16 half), VGPR v gives M = v + half*8
  int d = tn * 16 + l;
  float be  = b_emb[d];
  float wsd = Wsz[d];
  float bsd = bsz[d];
#pragma unroll
  for (int v = 0; v < 8; ++v) {
    int m = tm * 16 + half * 8 + v;
    float den = denom[m];
    float inv = 1.0f / (den + EPS_F);
    float pooled = inv * c[v] + (den * inv) * be;
    float upd = counts[m] * wsd + bsd;
    out[(size_t)m * DIM + d] = pooled * upd;
  }
}

extern "C" void kernel_launch(void* const* d_in, const int* in_sizes, int n_in,
                              void* d_out, int out_size, void* d_ws, size_t ws_size,
                              hipStream_t stream) {
  const float* x    = (const float*)d_in[0];
  const float* ax   = (const float*)d_in[1];
  const float* Wemb = (const float*)d_in[2];
  const float* bemb = (const float*)d_in[3];
  const float* Wsc  = (const float*)d_in[4];
  const float* bsc  = (const float*)d_in[5];
  const float* Wsz  = (const float*)d_in[6];
  const float* bsz  = (const float*)d_in[7];
  const int*   idx  = (const int*)d_in[8];

  float* ws      = (float*)d_ws;
  float* Y       = ws;                        // SEGS*DIM   (8 MB, L2-resident)
  float* scores  = Y + (size_t)SEGS * DIM;    // N_NODES
  float* seg_max = scores + N_NODES;          // SEGS
  float* denom   = seg_max + SEGS;            // SEGS
  float* counts  = denom + SEGS;              // SEGS
  float* out     = (float*)d_out;

  init_kernel<<<(SEGS * DIM + 255) / 256, 256, 0, stream>>>(Y, seg_max, denom, counts);
  scores_kernel<<<N_NODES / 8, 256, 0, stream>>>(ax, Wsc, bsc, idx, scores, seg_max, counts);
  accumulate_kernel<<<N_NODES, 128, 0, stream>>>(x, idx, scores, seg_max, denom, Y);
  wmma_gemm_epilogue<<<(SEGS / 16) * (DIM / 16) / 4, 128, 0, stream>>>(
      Y, Wemb, bemb, Wsz, bsz, denom, counts, out);
}